// ClassBlock_55078660603912
// MI455X (gfx1250) — compile-verified
//
#include <hip/hip_runtime.h>
#include <math.h>

// ---------------------------------------------------------------------------
// Shapes (fixed from the reference)
// B=16, NTOK=4097, C=1024, DG=256, DSTATE=1, DTRANK=16, HID=4096, RED=64
// Output = concat([cls_transformed, x[:,1:]]).  x[:,1:] copy dominates:
// 537 MB traffic -> ~23us @ 23.3 TB/s.  CLS path: ~0.3 GFLOP, negligible.
// All 16-row GEMMs use V_WMMA_F32_16X16X4_F32 (fp32-exact, M=16 tile).
// ---------------------------------------------------------------------------

typedef __attribute__((ext_vector_type(2))) float v2f;
typedef __attribute__((ext_vector_type(8))) float v8f;

#define NTOKC 4195328   // 4097*1024 floats per batch row of x / out

__device__ __forceinline__ float sigmoidf_(float x) { return 1.f / (1.f + expf(-x)); }
__device__ __forceinline__ float siluf_(float x)    { return x / (1.f + expf(-x)); }

__device__ __forceinline__ float apply_act(float v, int act) {
  switch (act) {
    case 1: return v > 0.f ? v : 0.f;                           // relu
    case 2: return sigmoidf_(v);                                // sigmoid
    case 3: return 0.5f * v * (1.f + erff(v * 0.70710678118654752f)); // exact gelu
  }
  return v;
}

// ------------------------- big passthrough copy ----------------------------
// out[:,1:,:] = x[:,1:,:]   (16 * 2^20 float4 per batch-row slice)
__global__ __launch_bounds__(256) void passthrough_kernel(const float4* __restrict__ x,
                                                          float4* __restrict__ out) {
  size_t i = (size_t)blockIdx.x * blockDim.x + threadIdx.x;  // < 16 * 2^20
  size_t b = i >> 20;
  size_t r = i & 1048575u;
  size_t off = b * (NTOKC / 4) + 256 + r;   // skip token 0 (1024 floats)
  out[off] = x[off];
}

// out[b,0,:]? no -- extract cls:  cls0[b*1024+c] = x[b*NTOKC + c]
__global__ void copycls_kernel(const float* __restrict__ x, float* __restrict__ cls0) {
  int idx = blockIdx.x * blockDim.x + threadIdx.x;  // 16384
  int b = idx >> 10, c = idx & 1023;
  cls0[idx] = x[(size_t)b * NTOKC + c];
}

// ------------------------------- layernorm ---------------------------------
// One block per row.  row = b*ngroups + g ; data at in + row*n ; w/b at g*n.
// Optional residual added after affine (residual indexed like out).
__global__ __launch_bounds__(256) void ln_kernel(const float* __restrict__ in,
                                                 const float* __restrict__ w,
                                                 const float* __restrict__ bi,
                                                 const float* __restrict__ residual,
                                                 float* __restrict__ out,
                                                 int n, int ngroups) {
  int row = blockIdx.x;
  int g = row % ngroups;
  const float* x = in + (size_t)row * n;
  __shared__ float s1[256], s2[256];
  float a = 0.f, q = 0.f;
  for (int i = threadIdx.x; i < n; i += 256) { float v = x[i]; a += v; q += v * v; }
  s1[threadIdx.x] = a; s2[threadIdx.x] = q;
  __syncthreads();
  for (int s = 128; s > 0; s >>= 1) {
    if ((int)threadIdx.x < s) { s1[threadIdx.x] += s1[threadIdx.x + s];
                                s2[threadIdx.x] += s2[threadIdx.x + s]; }
    __syncthreads();
  }
  float mean = s1[0] / n;
  float var  = s2[0] / n - mean * mean;
  float rstd = rsqrtf(var + 1e-5f);
  for (int i = threadIdx.x; i < n; i += 256) {
    float v = (x[i] - mean) * rstd * w[g * n + i] + bi[g * n + i];
    if (residual) v += residual[(size_t)row * n + i];
    out[(size_t)row * n + i] = v;
  }
}

// ------------------------- WMMA fp32 GEMM (M=16) ---------------------------
// C(16 x N) = act( A(16 x K) @ W(K x N) + bias ), one wave per 16-col tile.
// Fragment layouts per CDNA5 ISA 7.12.2 (32-bit A 16x4 / B 4x16 / D 16x16).
__global__ __launch_bounds__(32) void gemm16_wmma_kernel(
    const float* __restrict__ A, int lda,
    const float* __restrict__ W, int ldw,
    const float* __restrict__ bias,
    float* __restrict__ C, int ldc,
    int K, int act) {
  const int lane = threadIdx.x;
  const int half = lane >> 4;   // 0 or 1
  const int pos  = lane & 15;
  const int n0   = blockIdx.x << 4;
  v8f acc = {};
  const float* arow = A + (size_t)pos * lda + 2 * half;        // M=pos, K base 2*half
  const float* wcol = W + (size_t)(2 * half) * ldw + n0 + pos; // K base 2*half, N=pos
  for (int k = 0; k < K; k += 4) {
    v2f a, b;
    a.x = arow[k];
    a.y = arow[k + 1];
    b.x = wcol[(size_t)k * ldw];
    b.y = wcol[(size_t)(k + 1) * ldw];
    acc = __builtin_amdgcn_wmma_f32_16x16x4_f32(false, a, false, b,
                                                (short)0, acc, false, false);
  }
  int n = n0 + pos;
  float bv = bias ? bias[n] : 0.f;
#pragma unroll
  for (int j = 0; j < 8; ++j) {
    int m = j + 8 * half;
    C[(size_t)m * ldc + n] = apply_act(acc[j] + bv, act);
  }
}

// --------------------- ss2d elementwise pieces (L=1) -----------------------
// split in_proj result; 1x1 depthwise conv == center tap; silu both paths
__global__ void split_conv_kernel(const float* __restrict__ xz,
                                  const float* __restrict__ conv_w,
                                  const float* __restrict__ conv_b,
                                  float* __restrict__ xc, float* __restrict__ zs) {
  int idx = blockIdx.x * blockDim.x + threadIdx.x;  // 16384
  int b = idx >> 10, c = idx & 1023, g = c >> 8, d = c & 255;
  const float* xzg = xz + g * 8192 + b * 512;       // per-group 16x512 tile
  float xs = xzg[d];
  float z  = xzg[256 + d];
  float cw = conv_w[g * 2304 + d * 9 + 4];          // (4,256,3,3) center tap
  float t  = xs * cw + conv_b[g * 256 + d];
  xc[idx] = siluf_(t);
  zs[idx] = siluf_(z);
}

// x_dbl[b,g,e] = sum_d xc[b,g*256+d] * x_proj[g,d,e]   (e in 0..17)
__global__ void xproj_kernel(const float* __restrict__ xc,
                             const float* __restrict__ xpw,
                             float* __restrict__ xdbl) {
  int idx = blockIdx.x * blockDim.x + threadIdx.x;
  if (idx >= 16 * 4 * 18) return;
  int e = idx % 18, gb = idx / 18, g = gb % 4, b = gb / 4;
  const float* u  = xc + b * 1024 + g * 256;
  const float* wp = xpw + g * 4608 + e;
  float s = 0.f;
  for (int d = 0; d < 256; ++d) s += u[d] * wp[d * 18];
  xdbl[(b * 4 + g) * 18 + e] = s;
}

// delta = softplus(dts@dt_w + dt_b); scan(L=1,h0=0): y = delta*u*Bs*Cs + D*u
__global__ void deltay_kernel(const float* __restrict__ xc,
                              const float* __restrict__ xdbl,
                              const float* __restrict__ dt_w,
                              const float* __restrict__ dt_b,
                              const float* __restrict__ Dp,
                              float* __restrict__ ypre) {
  int idx = blockIdx.x * blockDim.x + threadIdx.x;  // 16384
  int b = idx >> 10, c = idx & 1023, g = c >> 8, d = c & 255;
  const float* xd = xdbl + (b * 4 + g) * 18;
  float s = dt_b[g * 256 + d];
  for (int r = 0; r < 16; ++r) s += xd[r] * dt_w[g * 4096 + r * 256 + d];
  float delta = (s > 20.f) ? s : log1pf(expf(s));
  float u = xc[idx];
  float Bs = xd[16], Cs = xd[17];
  ypre[idx] = delta * u * Bs * Cs + Dp[g * 256 + d] * u;
}

__global__ void mul_kernel(const float* __restrict__ a, const float* __restrict__ b,
                           float* __restrict__ o) {
  int idx = blockIdx.x * blockDim.x + threadIdx.x;
  o[idx] = a[idx] * b[idx];
}

// t = yproj * skip_scale * xn * se
__global__ void mix_kernel(const float* __restrict__ yproj, const float* __restrict__ skip,
                           const float* __restrict__ xn, const float* __restrict__ se,
                           float* __restrict__ o) {
  int idx = blockIdx.x * blockDim.x + threadIdx.x;
  o[idx] = yproj[idx] * skip[0] * xn[idx] * se[idx];
}

// out[b,0,:] = cls1 + fc2out
__global__ void final_kernel(const float* __restrict__ cls1, const float* __restrict__ h3,
                             float* __restrict__ out) {
  int idx = blockIdx.x * blockDim.x + threadIdx.x;  // 16384
  int b = idx >> 10, c = idx & 1023;
  out[(size_t)b * NTOKC + c] = cls1[idx] + h3[idx];
}

// ---------------------------------------------------------------------------
extern "C" void kernel_launch(void* const* d_in, const int* in_sizes, int n_in,
                              void* d_out, int out_size, void* d_ws, size_t ws_size,
                              hipStream_t stream) {
  (void)in_sizes; (void)n_in; (void)out_size; (void)ws_size;
  const float* x        = (const float*)d_in[0];
  const float* norm1_w  = (const float*)d_in[3];
  const float* norm1_b  = (const float*)d_in[4];
  const float* norm2_w  = (const float*)d_in[5];
  const float* norm2_b  = (const float*)d_in[6];
  const float* fc1_w    = (const float*)d_in[7];   // (1024,4096)
  const float* fc1_b    = (const float*)d_in[8];
  const float* fc2_w    = (const float*)d_in[9];   // (4096,1024)
  const float* fc2_b    = (const float*)d_in[10];
  const float* gm_w     = (const float*)d_in[11];
  const float* gm_b     = (const float*)d_in[12];
  const float* se1_w    = (const float*)d_in[13];  // (1024,64)
  const float* se1_b    = (const float*)d_in[14];
  const float* se2_w    = (const float*)d_in[15];  // (64,1024)
  const float* se2_b    = (const float*)d_in[16];
  const float* gmp_w    = (const float*)d_in[17];  // (1024,1024)
  const float* gmp_b    = (const float*)d_in[18];
  const float* skip     = (const float*)d_in[19];  // (1,)
  const float* in_proj  = (const float*)d_in[20];  // (4,256,512)
  const float* conv_w   = (const float*)d_in[21];  // (4,256,3,3)
  const float* conv_b   = (const float*)d_in[22];  // (4,256)
  const float* x_proj   = (const float*)d_in[23];  // (4,256,18)
  const float* dt_w     = (const float*)d_in[24];  // (4,16,256)
  const float* dt_b     = (const float*)d_in[25];  // (4,256)
  const float* Dp       = (const float*)d_in[27];  // (4,256)  (A_log dead: L=1)
  const float* onw      = (const float*)d_in[28];  // (4,256)
  const float* onb      = (const float*)d_in[29];  // (4,256)
  const float* out_proj = (const float*)d_in[30];  // (4,256,256)
  float* out = (float*)d_out;

  float* ws   = (float*)d_ws;
  float* cls0 = ws;            // 16384
  float* xn   = ws + 16384;    // 16384
  float* se_t = ws + 32768;    // 1024
  float* se   = ws + 34816;    // 16384
  float* xz   = ws + 51200;    // 4*16*512 = 32768
  float* xc   = ws + 83968;    // 16384
  float* zs   = ws + 100352;   // 16384
  float* xdbl = ws + 116736;   // 1152 (pad to 2048)
  float* ypre = ws + 118784;   // 16384
  float* yln  = ws + 135168;   // 16384
  float* ymul = ws + 151552;   // 16384
  float* ypj  = ws + 167936;   // 16384
  float* tmix = ws + 184320;   // 16384
  float* y3   = ws + 200704;   // 16384
  float* abuf = ws + 217088;   // 16384
  float* cls1 = ws + 233472;   // 16384
  float* hbuf = ws + 249856;   // 16384
  float* h2   = ws + 266240;   // 65536
  float* f2o  = ws + 331776;   // 16384

  // 1. passthrough copy: dominates runtime (bandwidth-bound)
  passthrough_kernel<<<65536, 256, 0, stream>>>((const float4*)x, (float4*)out);

  // 2. CLS path (16 rows)
  copycls_kernel<<<64, 256, 0, stream>>>(x, cls0);
  ln_kernel<<<16, 256, 0, stream>>>(cls0, gm_w, gm_b, nullptr, xn, 1024, 1);

  // SE: relu(xn@W1+b) -> sigmoid(@W2+b)
  gemm16_wmma_kernel<<<4,  32, 0, stream>>>(xn,  1024, se1_w, 64,   se1_b, se_t, 64,   1024, 1);
  gemm16_wmma_kernel<<<64, 32, 0, stream>>>(se_t, 64,  se2_w, 1024, se2_b, se,   1024, 64,   2);

  // per-group in_proj: (16x256)@(256x512)
  for (int g = 0; g < 4; ++g)
    gemm16_wmma_kernel<<<32, 32, 0, stream>>>(xn + g * 256, 1024,
                                              in_proj + g * 131072, 512,
                                              nullptr, xz + g * 8192, 512, 256, 0);
  split_conv_kernel<<<64, 256, 0, stream>>>(xz, conv_w, conv_b, xc, zs);
  xproj_kernel<<<9, 128, 0, stream>>>(xc, x_proj, xdbl);
  deltay_kernel<<<64, 256, 0, stream>>>(xc, xdbl, dt_w, dt_b, Dp, ypre);
  ln_kernel<<<64, 256, 0, stream>>>(ypre, onw, onb, nullptr, yln, 256, 4);  // group LN
  mul_kernel<<<64, 256, 0, stream>>>(yln, zs, ymul);
  for (int g = 0; g < 4; ++g)
    gemm16_wmma_kernel<<<16, 32, 0, stream>>>(ymul + g * 256, 1024,
                                              out_proj + g * 65536, 256,
                                              nullptr, ypj + g * 256, 1024, 256, 0);

  // gate/scale -> LN -> gm_proj -> residual LN
  mix_kernel<<<64, 256, 0, stream>>>(ypj, skip, xn, se, tmix);
  ln_kernel<<<16, 256, 0, stream>>>(tmix, gm_w, gm_b, nullptr, y3, 1024, 1);
  gemm16_wmma_kernel<<<64, 32, 0, stream>>>(y3, 1024, gmp_w, 1024, gmp_b, abuf, 1024, 1024, 0);
  ln_kernel<<<16, 256, 0, stream>>>(abuf, norm1_w, norm1_b, cls0, cls1, 1024, 1);

  // MLP: gelu fused into fc1 epilogue
  ln_kernel<<<16, 256, 0, stream>>>(cls1, norm2_w, norm2_b, nullptr, hbuf, 1024, 1);
  gemm16_wmma_kernel<<<256, 32, 0, stream>>>(hbuf, 1024, fc1_w, 4096, fc1_b, h2, 4096, 1024, 3);
  gemm16_wmma_kernel<<<64,  32, 0, stream>>>(h2, 4096, fc2_w, 1024, fc2_b, f2o, 1024, 4096, 0);
  final_kernel<<<64, 256, 0, stream>>>(cls1, f2o, out);
}